// LCNN_75136157876648
// MI455X (gfx1250) — compile-verified
//
#include <hip/hip_runtime.h>
#include <hip/hip_bf16.h>

// ---------------------------------------------------------------------------
// LCNN forward for MI455X (gfx1250, wave32).
// Pipeline: W(plaq+polyakov) -> U_PT -> [WMMA f32 16x16x4 omega-contraction]
//           -> alpha bilinear -> antiherm projection -> exp(iH)U -> W(EU).
// Workspace need: 16384 * 522 * 8 B ~= 68.5 MB in d_ws.
// ---------------------------------------------------------------------------

#define DEVFN static __device__ __forceinline__

typedef float v2f __attribute__((ext_vector_type(2)));
typedef float v8f __attribute__((ext_vector_type(8)));

struct c32 { float x, y; };                     // complex64 (re, im)
DEVFN c32 cadd(c32 a, c32 b){ return {a.x + b.x, a.y + b.y}; }
DEVFN c32 csub(c32 a, c32 b){ return {a.x - b.x, a.y - b.y}; }
DEVFN c32 cmul(c32 a, c32 b){ return {a.x*b.x - a.y*b.y, a.x*b.y + a.y*b.x}; }
DEVFN c32 cmulc(c32 a, c32 b){ return {a.x*b.x + a.y*b.y, a.y*b.x - a.x*b.y}; } // a*conj(b)
DEVFN c32 conjc(c32 a){ return {a.x, -a.y}; }
DEVFN c32 cscale(c32 a, float s){ return {a.x*s, a.y*s}; }

struct M3 { c32 e[9]; };

DEVFN M3 ldm(const c32* p){
  M3 r;
  #pragma unroll
  for (int i = 0; i < 9; ++i) r.e[i] = p[i];
  return r;
}
DEVFN void stm(c32* p, const M3& m){
  #pragma unroll
  for (int i = 0; i < 9; ++i) p[i] = m.e[i];
}
DEVFN M3 m3zero(){
  M3 r;
  #pragma unroll
  for (int i = 0; i < 9; ++i) r.e[i] = {0.f, 0.f};
  return r;
}
DEVFN M3 m3eye(){ M3 r = m3zero(); r.e[0] = r.e[4] = r.e[8] = {1.f, 0.f}; return r; }

DEVFN M3 mm(const M3& A, const M3& B){            // A @ B
  M3 C;
  #pragma unroll
  for (int r = 0; r < 3; ++r)
    #pragma unroll
    for (int c = 0; c < 3; ++c){
      c32 a = cmul(A.e[r*3+0], B.e[0*3+c]);
      a = cadd(a, cmul(A.e[r*3+1], B.e[1*3+c]));
      a = cadd(a, cmul(A.e[r*3+2], B.e[2*3+c]));
      C.e[r*3+c] = a;
    }
  return C;
}
DEVFN M3 mmh(const M3& A, const M3& B){           // A @ B^H
  M3 C;
  #pragma unroll
  for (int r = 0; r < 3; ++r)
    #pragma unroll
    for (int c = 0; c < 3; ++c){
      c32 a = cmulc(A.e[r*3+0], B.e[c*3+0]);
      a = cadd(a, cmulc(A.e[r*3+1], B.e[c*3+1]));
      a = cadd(a, cmulc(A.e[r*3+2], B.e[c*3+2]));
      C.e[r*3+c] = a;
    }
  return C;
}
DEVFN M3 adjm(const M3& A){
  M3 C;
  #pragma unroll
  for (int r = 0; r < 3; ++r)
    #pragma unroll
    for (int c = 0; c < 3; ++c) C.e[r*3+c] = conjc(A.e[c*3+r]);
  return C;
}

// Lattice constants: B=4 batches, 8^4 sites, D=4 directions, Nc=3, K=2.
#define NSITE 16384

DEVFN int shift_site(int s, int mu, int o){
  int cd[4] = { (s >> 9) & 7, (s >> 6) & 7, (s >> 3) & 7, s & 7 };
  int b = s >> 12;
  cd[mu] = (cd[mu] + o + 64) & 7;
  return (((b*8 + cd[0])*8 + cd[1])*8 + cd[2])*8 + cd[3];
}

// ---------------------------------------------------------------------------
// Kernel: W = [6 plaquettes ; 4 polyakov] per site.  (also used on EU -> out)
// ---------------------------------------------------------------------------
__global__ void lcnn_W_kernel(const c32* __restrict__ U, c32* __restrict__ W){
  int s = blockIdx.x * blockDim.x + threadIdx.x;
  if (s >= NSITE) return;
  __builtin_prefetch(U + (size_t)((s + 64) & (NSITE - 1)) * 36, 0, 3); // global_prefetch_b8

  M3 Um[4];
  #pragma unroll
  for (int mu = 0; mu < 4; ++mu) Um[mu] = ldm(U + ((size_t)s*4 + mu)*9);

  int ch = 0;
  #pragma unroll
  for (int mu = 0; mu < 4; ++mu)
    #pragma unroll
    for (int nu = mu + 1; nu < 4; ++nu){
      M3 b = ldm(U + ((size_t)shift_site(s, mu, 1)*4 + nu)*9);
      M3 c = ldm(U + ((size_t)shift_site(s, nu, 1)*4 + mu)*9);
      M3 t = mm(Um[mu], b);
      t = mmh(t, c);
      t = mmh(t, Um[nu]);
      stm(W + ((size_t)s*10 + ch)*9, t);
      ++ch;
    }
  #pragma unroll
  for (int mu = 0; mu < 4; ++mu){
    M3 P = Um[mu];
    for (int j = 1; j < 8; ++j)
      P = mm(P, ldm(U + ((size_t)shift_site(s, mu, j)*4 + mu)*9));
    stm(W + ((size_t)s*10 + ch)*9, P);
    ++ch;
  }
}

// ---------------------------------------------------------------------------
// Kernel: parallel transporters U_PT[s][mu][k], k=0..4 (k-2 = transport dist).
// ---------------------------------------------------------------------------
__global__ void lcnn_PT_kernel(const c32* __restrict__ U, c32* __restrict__ PT){
  int s = blockIdx.x * blockDim.x + threadIdx.x;
  if (s >= NSITE) return;
  #pragma unroll
  for (int mu = 0; mu < 4; ++mu){
    c32* base = PT + ((size_t)s*4 + mu)*5*9;
    stm(base + 2*9, m3eye());
    M3 T = ldm(U + ((size_t)s*4 + mu)*9);
    stm(base + 3*9, T);
    T = mm(T, ldm(U + ((size_t)shift_site(s, mu, 1)*4 + mu)*9));
    stm(base + 4*9, T);
    M3 Tb = adjm(ldm(U + ((size_t)shift_site(s, mu, -1)*4 + mu)*9));
    stm(base + 1*9, Tb);
    Tb = mm(Tb, adjm(ldm(U + ((size_t)shift_site(s, mu, -2)*4 + mu)*9)));
    stm(base + 0*9, Tb);
  }
}

// ---------------------------------------------------------------------------
// WMMA kernel: Wc[i] = sum_{j,m,k} omega[i,j,m,k] * (T W T^H).
// One wave per site, 2 sites/block (64 threads).
//   A = omega, pre-permuted into LDS as k-pairs x (16 re | 16 im) float2,
//   B = per-site Wp entries, staged as k-pairs x 18 cols (9 re | 9 im) float2,
//   D = 4x V_WMMA_F32_16X16X4_F32 accumulators, complex recombined via LDS.
// Inner loop: 4 ds_load_b64 + 4 v_wmma per k-step, no div/mod address math.
// ---------------------------------------------------------------------------
DEVFN int om_flat(int c, int i){        // combo c = m*50 + kmu*10 + j
  int j = c % 10, kmu = (c / 10) % 5, m = c / 50;
  return ((i*10 + j)*4 + m)*5 + kmu;
}

__global__ void lcnn_Wc_wmma_kernel(const c32* __restrict__ PT,
                                    const c32* __restrict__ W,
                                    const float* __restrict__ om, // interleaved c64
                                    c32* __restrict__ Wc){
  __shared__ float2 wp[2][100][18];    // [site][k-pair][entry col]  28.8 KB
  __shared__ float2 omp[100][32];      // [k-pair][16 re ch | 16 im ch] 25.6 KB
  const int tid  = threadIdx.x;
  const int w    = tid >> 5;
  const int lane = tid & 31;
  const int s    = blockIdx.x * 2 + w;
  const int rcol = lane & 15;
  const int hi   = lane >> 4;

  // ---- stage omega (whole block, once): A-fragment-ready layout ----
  for (int idx = tid; idx < 6400; idx += 64){
    int k = idx >> 5, col = idx & 31;
    int i = col & 15, part = col >> 4;
    ((float*)omp)[(k >> 1)*64 + col*2 + (k & 1)] = om[2*om_flat(k, i) + part];
  }

  // ---- stage Wp rows (per wave/site): k-pair packed ----
  for (int c = lane; c < 200; c += 32){
    int m = c / 50, kmu = (c / 10) % 5, j = c % 10;
    M3 T  = ldm(PT + (((size_t)s*4 + m)*5 + kmu)*9);
    int sn = shift_site(s, m, kmu - 2);
    M3 Wl = ldm(W + ((size_t)sn*10 + j)*9);
    M3 Sx = mmh(mm(T, Wl), T);
    float* fb = (float*)wp[w][c >> 1];
    #pragma unroll
    for (int e = 0; e < 9; ++e){
      fb[2*e + (c & 1)]       = Sx.e[e].x;
      fb[2*(9 + e) + (c & 1)] = Sx.e[e].y;
    }
  }
  __syncthreads();

  // ---- GEMM: 50 k-steps, 4 WMMAs each, unrolled x2 ----
  v8f accR0 = {0.f,0.f,0.f,0.f,0.f,0.f,0.f,0.f};
  v8f accR1 = accR0, accI0 = accR0, accI1 = accR0;
  const float msk = (rcol < 2) ? 1.f : 0.f;
  const int bcol = 16 + (rcol & 1);    // clamped in-bounds tile-1 column

  #pragma unroll 2
  for (int kt = 0; kt < 50; ++kt){
    int p = kt*2 + hi;
    float2 aRv = omp[p][rcol];
    float2 aIv = omp[p][16 + rcol];
    float2 b0v = wp[w][p][rcol];
    float2 b1v = wp[w][p][bcol];
    v2f aR = { aRv.x, aRv.y };
    v2f aI = { aIv.x, aIv.y };
    v2f b0 = { b0v.x, b0v.y };
    v2f b1 = { b1v.x * msk, b1v.y * msk };
    accR0 = __builtin_amdgcn_wmma_f32_16x16x4_f32(false, aR, false, b0, (short)0, accR0, false, false);
    accR1 = __builtin_amdgcn_wmma_f32_16x16x4_f32(false, aR, false, b1, (short)0, accR1, false, false);
    accI0 = __builtin_amdgcn_wmma_f32_16x16x4_f32(false, aI, false, b0, (short)0, accI0, false, false);
    accI1 = __builtin_amdgcn_wmma_f32_16x16x4_f32(false, aI, false, b1, (short)0, accI1, false, false);
  }
  __syncthreads();

  // ---- dump accumulators (C/D layout: row = v + 8*hi, col = lane%16) ----
  float* dump = (float*)wp[w];         // reuse as [32 rows][18 cols]
  #pragma unroll
  for (int v = 0; v < 8; ++v){
    int row = v + 8*hi;
    dump[row*18 + rcol] = accR0[v];
    if (rcol < 2) dump[row*18 + 16 + rcol] = accR1[v];
    dump[(16 + row)*18 + rcol] = accI0[v];
    if (rcol < 2) dump[(16 + row)*18 + 16 + rcol] = accI1[v];
  }
  __syncthreads();

  // ---- complex recombination:  Wc = (omR + i omI) @ (WpR + i WpI) ----
  for (int t = lane; t < 144; t += 32){
    int i = t / 9, e = t % 9;
    c32 v;
    v.x = dump[i*18 + e]       - dump[(16 + i)*18 + 9 + e];
    v.y = dump[i*18 + 9 + e]   + dump[(16 + i)*18 + e];
    Wc[((size_t)s*16 + i)*9 + e] = v;
  }
}

// ---------------------------------------------------------------------------
// alpha bilinear + trace gate + anti-hermitian traceless projection -> A[s][i]
// ---------------------------------------------------------------------------
__global__ void lcnn_alpha_kernel(const c32* __restrict__ alpha,
                                  const c32* __restrict__ Wc,
                                  c32* __restrict__ Aout){
  int t = blockIdx.x * blockDim.x + threadIdx.x;
  if (t >= NSITE * 8) return;
  int s = t >> 3, i = t & 7;

  M3 Wb = m3zero();
  for (int j = 0; j < 16; ++j){
    M3 R = m3zero();
    for (int k = 0; k < 16; ++k){
      c32 al = alpha[((size_t)i*16 + j)*16 + k];
      const c32* wk = Wc + ((size_t)s*16 + k)*9;
      #pragma unroll
      for (int e = 0; e < 9; ++e) R.e[e] = cadd(R.e[e], cmul(al, wk[e]));
    }
    M3 wj = ldm(Wc + ((size_t)s*16 + j)*9);
    M3 pr = mm(wj, R);
    #pragma unroll
    for (int e = 0; e < 9; ++e) Wb.e[e] = cadd(Wb.e[e], pr.e[e]);
  }
  float f = Wb.e[0].x + Wb.e[4].x + Wb.e[8].x;       // Re tr
  M3 Wa;
  #pragma unroll
  for (int e = 0; e < 9; ++e) Wa.e[e] = cscale(Wb.e[e], f);
  M3 A;
  #pragma unroll
  for (int r = 0; r < 3; ++r)
    #pragma unroll
    for (int c = 0; c < 3; ++c)
      A.e[r*3+c] = cscale(csub(Wa.e[r*3+c], conjc(Wa.e[c*3+r])), 0.5f);
  c32 tr = cadd(cadd(A.e[0], A.e[4]), A.e[8]);
  tr = cscale(tr, 1.f/3.f);
  A.e[0] = csub(A.e[0], tr); A.e[4] = csub(A.e[4], tr); A.e[8] = csub(A.e[8], tr);
  stm(Aout + ((size_t)s*8 + i)*9, A);
}

// ---------------------------------------------------------------------------
// H = i beta . A  (hermitian);  exp(iH) analytically (trig eigenvalues +
// Sylvester spectral projectors);  EU = exp(iH) @ U.
// ---------------------------------------------------------------------------
__global__ void lcnn_expU_kernel(const c32* __restrict__ beta,
                                 const c32* __restrict__ Ain,
                                 const c32* __restrict__ U,
                                 c32* __restrict__ EU){
  int t = blockIdx.x * blockDim.x + threadIdx.x;
  if (t >= NSITE * 4) return;
  int s = t >> 2, m = t & 3;

  M3 H = m3zero();
  for (int i = 0; i < 8; ++i){
    c32 bt = beta[m*8 + i];
    c32 ib = {-bt.y, bt.x};                           // i * beta
    const c32* a = Ain + ((size_t)s*8 + i)*9;
    #pragma unroll
    for (int e = 0; e < 9; ++e) H.e[e] = cadd(H.e[e], cmul(ib, a[e]));
  }
  M3 Hh;                                              // hermitize (numerics)
  #pragma unroll
  for (int r = 0; r < 3; ++r)
    #pragma unroll
    for (int c = 0; c < 3; ++c)
      Hh.e[r*3+c] = cscale(cadd(H.e[r*3+c], conjc(H.e[c*3+r])), 0.5f);

  float q = (Hh.e[0].x + Hh.e[4].x + Hh.e[8].x) / 3.f;
  M3 Bm = Hh;
  Bm.e[0].x -= q; Bm.e[4].x -= q; Bm.e[8].x -= q;
  float p2 = 0.f;
  #pragma unroll
  for (int e = 0; e < 9; ++e) p2 += Bm.e[e].x*Bm.e[e].x + Bm.e[e].y*Bm.e[e].y;
  float p = sqrtf(fmaxf(p2/6.f, 1e-30f));

  c32 d0 = cmul(Bm.e[4], Bm.e[8]); d0 = csub(d0, cmul(Bm.e[5], Bm.e[7]));
  c32 d1 = cmul(Bm.e[3], Bm.e[8]); d1 = csub(d1, cmul(Bm.e[5], Bm.e[6]));
  c32 d2 = cmul(Bm.e[3], Bm.e[7]); d2 = csub(d2, cmul(Bm.e[4], Bm.e[6]));
  c32 det = csub(cmul(Bm.e[0], d0), cmul(Bm.e[1], d1));
  det = cadd(det, cmul(Bm.e[2], d2));
  float rr = det.x / (2.f*p*p*p);
  rr = fminf(1.f, fmaxf(-1.f, rr));
  float phi = acosf(rr) / 3.f;
  float lam[3];
  lam[0] = q + 2.f*p*__cosf(phi);
  lam[2] = q + 2.f*p*__cosf(phi + 2.09439510239f);
  lam[1] = 3.f*q - lam[0] - lam[2];

  M3 E = m3zero();
  #pragma unroll
  for (int a = 0; a < 3; ++a){
    int b = (a + 1) % 3, c = (a + 2) % 3;
    M3 Mb = Hh; Mb.e[0].x -= lam[b]; Mb.e[4].x -= lam[b]; Mb.e[8].x -= lam[b];
    M3 Mc = Hh; Mc.e[0].x -= lam[c]; Mc.e[4].x -= lam[c]; Mc.e[8].x -= lam[c];
    M3 P = mm(Mb, Mc);
    float den = (lam[a] - lam[b]) * (lam[a] - lam[c]);
    if (fabsf(den) < 1e-12f) den = copysignf(1e-12f, den);
    c32 ph = { __cosf(lam[a]) / den, __sinf(lam[a]) / den };   // e^{i lam}/den
    #pragma unroll
    for (int e = 0; e < 9; ++e) E.e[e] = cadd(E.e[e], cmul(ph, P.e[e]));
  }
  M3 Um = ldm(U + ((size_t)s*4 + m)*9);
  stm(EU + ((size_t)s*4 + m)*9, mm(E, Um));
}

// ---------------------------------------------------------------------------
extern "C" void kernel_launch(void* const* d_in, const int* in_sizes, int n_in,
                              void* d_out, int out_size, void* d_ws, size_t ws_size,
                              hipStream_t stream){
  const c32*   U     = (const c32*)d_in[0];
  const float* omega = (const float*)d_in[1];
  const c32*   alpha = (const c32*)d_in[2];
  const c32*   beta  = (const c32*)d_in[3];
  (void)in_sizes; (void)n_in; (void)out_size; (void)ws_size;   // K=2 hardcoded

  c32* W    = (c32*)d_ws;                 // NSITE * 90
  c32* PT   = W    + (size_t)NSITE *  90; // NSITE * 180
  c32* Wc   = PT   + (size_t)NSITE * 180; // NSITE * 144
  c32* Aout = Wc   + (size_t)NSITE * 144; // NSITE * 72
  c32* EU   = Aout + (size_t)NSITE *  72; // NSITE * 36

  lcnn_W_kernel      <<<NSITE/256,     256, 0, stream>>>(U, W);
  lcnn_PT_kernel     <<<NSITE/256,     256, 0, stream>>>(U, PT);
  lcnn_Wc_wmma_kernel<<<NSITE/2,        64, 0, stream>>>(PT, W, omega, Wc);
  lcnn_alpha_kernel  <<<NSITE*8/256,   256, 0, stream>>>(alpha, Wc, Aout);
  lcnn_expU_kernel   <<<NSITE*4/256,   256, 0, stream>>>(beta, Aout, U, EU);
  lcnn_W_kernel      <<<NSITE/256,     256, 0, stream>>>(EU, (c32*)d_out);
}